// GraphConvolution_48679159332988
// MI455X (gfx1250) — compile-verified
//
#include <hip/hip_runtime.h>

typedef __attribute__((ext_vector_type(2))) float v2f;
typedef __attribute__((ext_vector_type(8))) float v8f;

#define D 512            // D_IN == D_OUT == 512
#define TILE_M 16

// ---------------------------------------------------------------------------
// Async global -> LDS copy (CDNA5 GLOBAL_LOAD_ASYNC_TO_LDS_B128, GV mode).
// Each active lane copies 16 bytes memory->LDS without touching VGPR data.
// Tracked by ASYNCcnt; wait with s_wait_asynccnt before reading the LDS.
// ---------------------------------------------------------------------------
__device__ __forceinline__ void async_copy_b128(unsigned ldsOff, const void* gptr)
{
    asm volatile("global_load_async_to_lds_b128 %0, %1, off"
                 :
                 : "v"(ldsOff), "v"(gptr)
                 : "memory");
}

__device__ __forceinline__ void wait_async0()
{
    asm volatile("s_wait_asynccnt 0x0" ::: "memory");
}

// ---------------------------------------------------------------------------
// Kernel 1: support = X @ W using V_WMMA_F32_16X16X4_F32 (fp32 matrix pipe)
// Block = 256 threads = 8 waves. Block b owns rows [16b, 16b+16) of X.
// Wave w owns output columns [64w, 64w+64): 4 WMMA tiles sharing one A frag.
// ---------------------------------------------------------------------------
__global__ __launch_bounds__(256) void gcn_gemm_wmma(const float* __restrict__ X,
                                                     const float* __restrict__ W,
                                                     float* __restrict__ support)
{
    __shared__ __align__(16) float As[TILE_M * D];   // 32 KB A panel (full K)

    const int tid   = threadIdx.x;
    const int wave  = tid >> 5;
    const int lane  = tid & 31;
    const int mTile = blockIdx.x;                    // 0 .. 3124

    // Async-stage 16 contiguous rows of X (one contiguous 32 KB chunk) to LDS.
    // Low 32 bits of the generic-cast shared pointer = LDS byte offset.
    const float4*  gsrc    = (const float4*)(X + (size_t)mTile * TILE_M * D);
    const unsigned ldsBase = (unsigned)(size_t)(void*)As;
    #pragma unroll
    for (int i = 0; i < 8; ++i)
        async_copy_b128(ldsBase + (unsigned)(i * 256 + tid) * 16,
                        gsrc + i * 256 + tid);
    wait_async0();
    __syncthreads();

    const int g     = lane >> 4;      // half-wave: selects K pair / row half
    const int hl    = lane & 15;      // row (A) / col (B,C) within tile
    const int nBase = wave * 64;      // 8 waves cover all 512 columns

    v8f acc0 = {}, acc1 = {}, acc2 = {}, acc3 = {};

    #pragma unroll 2
    for (int k0 = 0; k0 < D; k0 += 4) {
        const int ka = k0 + 2 * g;
        // A 16x4 fragment: lane holds A[hl][ka], A[hl][ka+1]  (ISA layout)
        v2f a;
        a.x = As[hl * D + ka];
        a.y = As[hl * D + ka + 1];

        // B 4x16 fragments: lane holds W[ka][n], W[ka+1][n]
        const float* w0 = W + (size_t)ka * D + nBase + hl;
        const float* w1 = w0 + D;
        v2f b0, b1, b2, b3;
        b0.x = w0[0];   b0.y = w1[0];
        b1.x = w0[16];  b1.y = w1[16];
        b2.x = w0[32];  b2.y = w1[32];
        b3.x = w0[48];  b3.y = w1[48];

        acc0 = __builtin_amdgcn_wmma_f32_16x16x4_f32(false, a, false, b0, (short)0, acc0, false, false);
        acc1 = __builtin_amdgcn_wmma_f32_16x16x4_f32(false, a, false, b1, (short)0, acc1, false, false);
        acc2 = __builtin_amdgcn_wmma_f32_16x16x4_f32(false, a, false, b2, (short)0, acc2, false, false);
        acc3 = __builtin_amdgcn_wmma_f32_16x16x4_f32(false, a, false, b3, (short)0, acc3, false, false);
    }

    // C/D 16x16 layout: VGPR v -> row v + 8*g, col hl.
    float* orow = support + (size_t)mTile * TILE_M * D + nBase;
    #pragma unroll
    for (int v = 0; v < 8; ++v) {
        float* p = orow + (size_t)(v + 8 * g) * D + hl;
        p[0]  = acc0[v];
        p[16] = acc1[v];
        p[32] = acc2[v];
        p[48] = acc3[v];
    }
}

// ---------------------------------------------------------------------------
// Kernel 2: out[i][:] = bias[:]  (broadcast init; segment-sum lands on top)
// One float4 per thread; bias (2 KB) is L2/L0 resident.
// ---------------------------------------------------------------------------
__global__ __launch_bounds__(256) void gcn_bias_init(const float* __restrict__ bias,
                                                     float* __restrict__ out,
                                                     int totalVec4)
{
    int i = blockIdx.x * 256 + threadIdx.x;        // over N_NODES * 128 float4s
    if (i >= totalVec4) return;
    float4 b = ((const float4*)bias)[i & 127];     // column chunk = i % 128
    ((float4*)out)[i] = b;
}

// ---------------------------------------------------------------------------
// Kernel 3: edge scatter. 128 threads/edge, float4 gather of support[col[e]],
// hardware fp32 atomic adds into out[row[e]]. Atomics resolve in L2.
// ---------------------------------------------------------------------------
__global__ __launch_bounds__(256) void gcn_spmm_scatter(const float* __restrict__ support,
                                                        const int* __restrict__ row,
                                                        const int* __restrict__ col,
                                                        const float* __restrict__ ew,
                                                        float* __restrict__ out,
                                                        int E)
{
    const int e = blockIdx.x * 2 + (threadIdx.x >> 7);   // 2 edges per block
    if (e >= E) return;
    const int t = threadIdx.x & 127;                     // 128 float4 chunks/row

    const int   r = row[e];
    const int   c = col[e];
    const float w = ew[e];

    const float4* src = (const float4*)(support + (size_t)c * D);
    __builtin_prefetch(src + t, 0, 0);                   // global_prefetch_b8
    float4 v = src[t];

    float* dst = out + (size_t)r * D + t * 4;
    unsafeAtomicAdd(dst + 0, w * v.x);                   // global_atomic_add_f32
    unsafeAtomicAdd(dst + 1, w * v.y);
    unsafeAtomicAdd(dst + 2, w * v.z);
    unsafeAtomicAdd(dst + 3, w * v.w);
}

// ---------------------------------------------------------------------------
extern "C" void kernel_launch(void* const* d_in, const int* in_sizes, int n_in,
                              void* d_out, int out_size, void* d_ws, size_t ws_size,
                              hipStream_t stream)
{
    const float* x    = (const float*)d_in[0];   // [N, 512]
    const float* w    = (const float*)d_in[1];   // [512, 512]
    const float* bias = (const float*)d_in[2];   // [512]
    const int*   row  = (const int*)  d_in[3];   // [E]
    const int*   col  = (const int*)  d_in[4];   // [E]
    const float* ew   = (const float*)d_in[5];   // [E]

    float* out     = (float*)d_out;
    float* support = (float*)d_ws;               // 50000*512*4 = 102.4 MB scratch

    const int nNodes = in_sizes[0] / D;          // 50000 (divisible by 16)
    const int E      = in_sizes[3];              // 1.6M

    // 1) support = X @ W  (fp32 WMMA)
    gcn_gemm_wmma<<<nNodes / TILE_M, 256, 0, stream>>>(x, w, support);

    // 2) out = broadcast(bias)
    const int totalVec4 = nNodes * (D / 4);
    gcn_bias_init<<<(totalVec4 + 255) / 256, 256, 0, stream>>>(bias, out, totalVec4);

    // 3) out[row[e]] += ew[e] * support[col[e]]
    gcn_spmm_scatter<<<(E + 1) / 2, 256, 0, stream>>>(support, row, col, ew, out, E);
}